// ESM2VAEModel_65721589563717
// MI455X (gfx1250) — compile-verified
//
#include <hip/hip_runtime.h>

typedef __attribute__((ext_vector_type(2))) float v2f;
typedef __attribute__((ext_vector_type(4))) float v4f;
typedef __attribute__((ext_vector_type(8))) float v8f;

namespace {
constexpr int kD       = 480;   // feature dim
constexpr int kHStride = 484;   // padded LDS row stride (floats) -> conflict-free
constexpr int kZW      = 40;    // LDS z-buffer width (covers cols 0..33)
constexpr int kMBlk    = 64;    // rows per workgroup (2 waves x 32 rows)
constexpr int kRPW     = 32;    // rows per wave (2 WMMA row-tiles, shared B frags)
constexpr int kPanel   = 96;    // output columns per accumulation pass
constexpr int kNT      = kPanel / 16;  // 6 WMMA col-tiles per panel
}

__global__ __launch_bounds__(64)
void esm2vae_fused(const float* __restrict__ X,
                   const float* __restrict__ W1, const float* __restrict__ b1,
                   const float* __restrict__ W2, const float* __restrict__ b2,
                   float* __restrict__ outX,   float* __restrict__ outZ,
                   float* __restrict__ outZ34, float* __restrict__ outB34)
{
    __shared__ float Hs[kMBlk * kHStride];   // relu(X@W1+b1) block, fp32
    __shared__ float zbuf[kMBlk * kZW];      // z columns 0..39 for epilogue

    const int  tid   = threadIdx.x;          // 0..63
    const int  wave  = tid >> 5;             // 0..1
    const int  lane  = tid & 31;
    const int  l16   = lane & 15;
    const int  lhalf = lane >> 4;            // 0: lanes 0-15, 1: lanes 16-31
    const long rowBlock = (long)blockIdx.x * kMBlk;
    const int  waveRow  = wave * kRPW;             // first row (in block) of wave
    const long gRow0    = rowBlock + waveRow;      // global first row of wave

    // ---- out0: passthrough copy (b128, non-temporal stores keep L2 for W/X) ----
    {
        const v4f* X4 = (const v4f*)X;
        v4f*       O4 = (v4f*)outX;
        const long base = rowBlock * (kD / 4);
        for (int i = tid; i < kMBlk * (kD / 4); i += 64) {
            v4f v = X4[base + i];                    // also warms L2 for A-frag reads
            __builtin_nontemporal_store(v, &O4[base + i]);
        }
    }

    // ---------- GEMM1: Hs = relu(X @ W1 + b1) ----------
    for (int p = 0; p < kD / kPanel; ++p) {
        const int c0 = p * kPanel;
        v8f acc[2][kNT];
#pragma unroll
        for (int t = 0; t < kNT; ++t) {
            const float bv = b1[c0 + 16 * t + l16];  // bias depends on column only
            v8f a;
#pragma unroll
            for (int r = 0; r < 8; ++r) a[r] = bv;
            acc[0][t] = a;
            acc[1][t] = a;
        }
        for (int k = 0; k < kD; k += 4) {
            // prefetch W1 panel rows 16 k-steps ahead into near cache (WGP scope);
            // uniform wrap keeps the address in-bounds without an inner-loop branch
            {
                const int pk = (k + 16 < kD) ? (k + 16) : (k + 16 - kD);
                const float* pf = W1 + (long)(pk + (lane >> 3)) * kD
                                     + c0 + (lane & 7) * 12;
                __builtin_prefetch((const void*)pf, 0, 3);
            }
            // A frags (16x4 f32): two row-tiles; lane half selects K pair {0,1}/{2,3}
            v2f af0 = *(const v2f*)(X + (gRow0      + l16) * kD + k + 2 * lhalf);
            v2f af1 = *(const v2f*)(X + (gRow0 + 16 + l16) * kD + k + 2 * lhalf);
#pragma unroll
            for (int t = 0; t < kNT; ++t) {
                const int col = c0 + 16 * t + l16;
                v2f bfrag;
                bfrag.x = W1[(long)(k + 2 * lhalf    ) * kD + col];
                bfrag.y = W1[(long)(k + 2 * lhalf + 1) * kD + col];
                acc[0][t] = __builtin_amdgcn_wmma_f32_16x16x4_f32(
                    false, af0, false, bfrag, (short)0, acc[0][t], false, false);
                acc[1][t] = __builtin_amdgcn_wmma_f32_16x16x4_f32(
                    false, af1, false, bfrag, (short)0, acc[1][t], false, false);
            }
        }
#pragma unroll
        for (int h = 0; h < 2; ++h) {
#pragma unroll
            for (int t = 0; t < kNT; ++t) {
                const int col = c0 + 16 * t + l16;
#pragma unroll
                for (int r = 0; r < 8; ++r) {
                    const int m = r + 8 * lhalf;     // C/D layout: VGPR r -> row r / r+8
                    float v = acc[h][t][r];
                    Hs[(waveRow + 16 * h + m) * kHStride + col] = v > 0.f ? v : 0.f;
                }
            }
        }
    }
    __syncthreads();

    // ---------- GEMM2: z = Hs @ W2 + b2 ----------
    for (int p = 0; p < kD / kPanel; ++p) {
        const int c0 = p * kPanel;
        v8f acc[2][kNT];
#pragma unroll
        for (int t = 0; t < kNT; ++t) {
            const float bv = b2[c0 + 16 * t + l16];
            v8f a;
#pragma unroll
            for (int r = 0; r < 8; ++r) a[r] = bv;
            acc[0][t] = a;
            acc[1][t] = a;
        }
        for (int k = 0; k < kD; k += 4) {
            {
                const int pk = (k + 16 < kD) ? (k + 16) : (k + 16 - kD);
                const float* pf = W2 + (long)(pk + (lane >> 3)) * kD
                                     + c0 + (lane & 7) * 12;
                __builtin_prefetch((const void*)pf, 0, 3);
            }
            v2f af0 = *(const v2f*)(&Hs[(waveRow      + l16) * kHStride + k + 2 * lhalf]);
            v2f af1 = *(const v2f*)(&Hs[(waveRow + 16 + l16) * kHStride + k + 2 * lhalf]);
#pragma unroll
            for (int t = 0; t < kNT; ++t) {
                const int col = c0 + 16 * t + l16;
                v2f bfrag;
                bfrag.x = W2[(long)(k + 2 * lhalf    ) * kD + col];
                bfrag.y = W2[(long)(k + 2 * lhalf + 1) * kD + col];
                acc[0][t] = __builtin_amdgcn_wmma_f32_16x16x4_f32(
                    false, af0, false, bfrag, (short)0, acc[0][t], false, false);
                acc[1][t] = __builtin_amdgcn_wmma_f32_16x16x4_f32(
                    false, af1, false, bfrag, (short)0, acc[1][t], false, false);
            }
        }
#pragma unroll
        for (int h = 0; h < 2; ++h) {
#pragma unroll
            for (int t = 0; t < kNT; ++t) {
                const int col = c0 + 16 * t + l16;
#pragma unroll
                for (int r = 0; r < 8; ++r) {
                    const int  m    = r + 8 * lhalf;
                    const long grow = gRow0 + 16 * h + m;
                    const float v = acc[h][t][r];
                    __builtin_nontemporal_store(v, &outZ[grow * kD + col]);
                    if (col < kZW) zbuf[(waveRow + 16 * h + m) * kZW + col] = v;
                }
            }
        }
    }
    __syncthreads();

    // ---------- epilogue: z[:, :34] and segment one-hot(argmax) ----------
    {
        const long grow = rowBlock + tid;        // 64 threads, one row each
        float zr[34];
#pragma unroll
        for (int c = 0; c < 34; ++c) zr[c] = zbuf[tid * kZW + c];

        int i1 = 0; { float m = zr[1];  for (int c = 1; c < 8;  ++c) if (zr[1  + c] > m) { m = zr[1  + c]; i1 = c; } }
        int i2 = 0; { float m = zr[9];  for (int c = 1; c < 3;  ++c) if (zr[9  + c] > m) { m = zr[9  + c]; i2 = c; } }
        int i3 = 0; { float m = zr[12]; for (int c = 1; c < 20; ++c) if (zr[12 + c] > m) { m = zr[12 + c]; i3 = c; } }

#pragma unroll
        for (int c = 0; c < 34; ++c) {
            outZ34[grow * 34 + c] = zr[c];
            float bv;
            if      (c >= 1  && c < 9 ) bv = (c - 1  == i1) ? 1.f : 0.f;
            else if (c >= 9  && c < 12) bv = (c - 9  == i2) ? 1.f : 0.f;
            else if (c >= 12 && c < 32) bv = (c - 12 == i3) ? 1.f : 0.f;
            else                        bv = zr[c];
            outB34[grow * 34 + c] = bv;
        }
    }
}

extern "C" void kernel_launch(void* const* d_in, const int* in_sizes, int n_in,
                              void* d_out, int out_size, void* d_ws, size_t ws_size,
                              hipStream_t stream) {
    (void)n_in; (void)out_size; (void)d_ws; (void)ws_size;
    const float* X  = (const float*)d_in[0];
    const float* W1 = (const float*)d_in[1];
    const float* b1 = (const float*)d_in[2];
    const float* W2 = (const float*)d_in[3];
    const float* b2 = (const float*)d_in[4];

    const long N = (long)in_sizes[0] / kD;   // 131072

    float* out    = (float*)d_out;
    float* outX   = out;                      // valid_embeddings passthrough
    float* outZ   = out + N * (long)kD;       // z
    float* outZ34 = out + 2 * N * (long)kD;   // z[:, :34]
    float* outB34 = outZ34 + N * 34;          // binary[:, :34]

    dim3 grid((unsigned)(N / kMBlk));
    dim3 block(64);
    esm2vae_fused<<<grid, block, 0, stream>>>(X, W1, b1, W2, b2,
                                              outX, outZ, outZ34, outB34);
}